// MambaBlock_858993459381
// MI455X (gfx1250) — compile-verified
//
#include <hip/hip_runtime.h>
#include <hip/hip_bf16.h>
#include <math.h>

// ---------------------------------------------------------------------------
// Mamba block forward for MI455X (gfx1250), f32 via V_WMMA_F32_16X16X4_F32.
//
//   B=2, L=1024, D_MODEL=1024, D_INNER=2048, D_STATE=16, D_CONV=4, DT_RANK=64
//
//   1) xz   = x @ W_in                 (2048 x 1024 x 4096)   [WMMA 32x64/wave]
//   2) u_c  = silu(causal_conv(u)+b)   elementwise
//   3) xdbl = u_c @ W_x                (2048 x 2048 x 96)     [WMMA 32x96/wave]
//   4) dt   = softplus(dtl@W_dt + b)   (2048 x 64 x 2048)     [WMMA + epilogue]
//   5) scan: 16 lanes per (b,d) channel (one per state), shfl-xor reduce,
//      fused with  y += u_c*D ;  y *= silu(z)
//   6) out  = y @ W_out                (2048 x 2048 x 1024)   [WMMA 32x64/wave]
// ---------------------------------------------------------------------------

#define BATCH   2
#define SEQ     1024
#define D_MODEL 1024
#define D_INNER 2048
#define D_STATE 16
#define D_CONV  4
#define DT_RANK 64
#define XZ_N    (2 * D_INNER)            // 4096
#define XD_N    (DT_RANK + 2 * D_STATE)  // 96
#define MROWS   (BATCH * SEQ)            // 2048

typedef float v2f __attribute__((ext_vector_type(2)));
typedef float v8f __attribute__((ext_vector_type(8)));

// ---------------------------------------------------------------------------
// f32 WMMA GEMM:  C[M x N] = A[M x K(lda)] * B[K x N(ldb)]
// One wave computes a (16*MTILE) x (16*NTILE) output tile, K in steps of 4.
// A fragments reused across NTILE, B fragments reused across MTILE.
// EPI: 0 = none, 1 = softplus(v + bias[col]).
// All branching is wave-uniform -> EXEC all ones at every WMMA (ISA req).
// ---------------------------------------------------------------------------
template <int MTILE, int NTILE, int EPI>
__global__ __launch_bounds__(256)
void gemm_wmma_f32(const float* __restrict__ A, const float* __restrict__ B,
                   float* __restrict__ C, const float* __restrict__ bias,
                   int M, int N, int K, int lda, int ldb, int ldc, int n_groups)
{
    const int wave   = blockIdx.x * (blockDim.x >> 5) + (threadIdx.x >> 5);
    const int lane   = threadIdx.x & 31;
    const int m_tile = wave / n_groups;
    const int n_grp  = wave % n_groups;
    if (m_tile * (16 * MTILE) >= M) return;        // wave-uniform guard

    const int half = lane >> 4;                    // 0: K+0/1, 1: K+2/3
    const int l16  = lane & 15;

    v8f acc[MTILE][NTILE];
#pragma unroll
    for (int mi = 0; mi < MTILE; ++mi)
#pragma unroll
        for (int j = 0; j < NTILE; ++j) acc[mi][j] = (v8f){};

    for (int k = 0; k < K; k += 4) {
        const int ka = k + 2 * half;
        // A fragments: 2 consecutive f32 of one row (ISA 16x4 f32 layout)
        v2f a[MTILE];
#pragma unroll
        for (int mi = 0; mi < MTILE; ++mi) {
            const size_t arow = (size_t)(m_tile * (16 * MTILE) + mi * 16 + l16);
            a[mi].x = A[arow * lda + ka];
            a[mi].y = A[arow * lda + ka + 1];
        }
#pragma unroll
        for (int j = 0; j < NTILE; ++j) {
            const int col = n_grp * (16 * NTILE) + j * 16 + l16;
            v2f b;
            b.x = B[(size_t)ka       * ldb + col];
            b.y = B[(size_t)(ka + 1) * ldb + col];
#pragma unroll
            for (int mi = 0; mi < MTILE; ++mi) {
                // 8 args: (neg_a, A, neg_b, B, c_mod, C, reuse_a, reuse_b)
                acc[mi][j] = __builtin_amdgcn_wmma_f32_16x16x4_f32(
                                 false, a[mi], false, b, (short)0, acc[mi][j],
                                 false, false);
            }
        }
    }

#pragma unroll
    for (int mi = 0; mi < MTILE; ++mi) {
#pragma unroll
        for (int j = 0; j < NTILE; ++j) {
            const int col = n_grp * (16 * NTILE) + j * 16 + l16;
            float bv = 0.0f;
            if (EPI == 1) bv = bias[col];
#pragma unroll
            for (int i = 0; i < 8; ++i) {
                const size_t row =
                    (size_t)(m_tile * (16 * MTILE) + mi * 16 + half * 8 + i);
                float v = acc[mi][j][i];
                if (EPI == 1) {                    // softplus(v + bias)
                    float t = v + bv;
                    v = (t > 20.0f) ? t : log1pf(__expf(t));
                }
                C[row * ldc + col] = v;
            }
        }
    }
}

// ---------------------------------------------------------------------------
// Depthwise causal conv (width 4) + bias + SiLU over u = xz[..., :D_INNER]
// ---------------------------------------------------------------------------
__global__ __launch_bounds__(256)
void conv_silu_kernel(const float* __restrict__ xz,
                      const float* __restrict__ conv_w,
                      const float* __restrict__ conv_b,
                      float* __restrict__ u_c)
{
    const int idx = blockIdx.x * blockDim.x + threadIdx.x;
    if (idx >= BATCH * SEQ * D_INNER) return;
    const int d = idx % D_INNER;
    const int l = (idx / D_INNER) % SEQ;
    const int b = idx / (D_INNER * SEQ);

    float acc = conv_b[d];
#pragma unroll
    for (int k = 0; k < D_CONV; ++k) {
        const int ll = l - (D_CONV - 1) + k;
        if (ll >= 0)
            acc += xz[((size_t)(b * SEQ + ll)) * XZ_N + d] * conv_w[d * D_CONV + k];
    }
    u_c[idx] = acc / (1.0f + __expf(-acc));        // silu
}

// ---------------------------------------------------------------------------
// Selective scan, state-parallel: 16 lanes per (b,d) channel (one per state),
// 2 channels per wave. Per step each lane does 1 exp + 2 fma, then a 4-step
// shfl-xor butterfly reduces y = sum_n h[n]*C[n] across the 16 lanes.
// Fuses y += u*D and y *= silu(z). 4096 channels -> 2048 waves.
// ---------------------------------------------------------------------------
__global__ __launch_bounds__(256)
void scan_kernel(const float* __restrict__ xz,     // z gate
                 const float* __restrict__ u_c,
                 const float* __restrict__ x_dbl,  // [., 96]: [64..79]=B, [80..95]=C
                 const float* __restrict__ dt,
                 const float* __restrict__ A_log,
                 const float* __restrict__ Dvec,
                 float* __restrict__ ybuf)
{
    const int tid  = blockIdx.x * blockDim.x + threadIdx.x;
    const int chan = tid >> 4;                     // one channel per 16 lanes
    const int n    = tid & 15;                     // state index
    if (chan >= BATCH * D_INNER) return;
    const int d = chan % D_INNER;
    const int b = chan / D_INNER;

    const float An = -__expf(A_log[d * D_STATE + n]);
    const float Dd = Dvec[d];
    float h = 0.0f;

    for (int l = 0; l < SEQ; ++l) {
        const size_t rl = (size_t)(b * SEQ + l);
        const float u   = u_c[rl * D_INNER + d];   // broadcast across 16 lanes
        const float dtv = dt [rl * D_INNER + d];
        const float Bn  = x_dbl[rl * XD_N + DT_RANK + n];           // coalesced
        const float Cn  = x_dbl[rl * XD_N + DT_RANK + D_STATE + n]; // coalesced

        h = __expf(dtv * An) * h + (dtv * u) * Bn;
        float p = h * Cn;
        // butterfly reduce across the 16 state lanes (ds-permute datapath)
        p += __shfl_xor(p, 1, 16);
        p += __shfl_xor(p, 2, 16);
        p += __shfl_xor(p, 4, 16);
        p += __shfl_xor(p, 8, 16);

        if (n == 0) {
            float y = p + u * Dd;
            const float z = xz[rl * XZ_N + D_INNER + d];
            y *= z / (1.0f + __expf(-z));          // silu(z) gate
            ybuf[rl * D_INNER + d] = y;
        }
    }
}

// ---------------------------------------------------------------------------
// Launcher
// ---------------------------------------------------------------------------
static inline int gemm_grid(int M, int n_groups)
{
    const int waves = (M / 32) * n_groups;         // MTILE = 2
    return (waves + 7) / 8;                        // 8 waves / 256-thread block
}

extern "C" void kernel_launch(void* const* d_in, const int* in_sizes, int n_in,
                              void* d_out, int out_size, void* d_ws, size_t ws_size,
                              hipStream_t stream)
{
    const float* x      = (const float*)d_in[0];
    const float* W_in   = (const float*)d_in[1];
    const float* conv_w = (const float*)d_in[2];
    const float* conv_b = (const float*)d_in[3];
    const float* W_x    = (const float*)d_in[4];
    const float* W_dt   = (const float*)d_in[5];
    const float* b_dt   = (const float*)d_in[6];
    const float* A_log  = (const float*)d_in[7];
    const float* Dvec   = (const float*)d_in[8];
    const float* W_out  = (const float*)d_in[9];
    float* out          = (float*)d_out;

    // Workspace layout (floats)
    char* ws = (char*)d_ws;
    float* xz    = (float*)ws;                                                    // 2048 x 4096
    float* u_c   = (float*)(ws + (size_t)MROWS * XZ_N * 4);                       // 2048 x 2048
    float* x_dbl = (float*)(ws + (size_t)MROWS * (XZ_N + D_INNER) * 4);           // 2048 x 96
    float* dt    = (float*)(ws + (size_t)MROWS * (XZ_N + D_INNER + XD_N) * 4);    // 2048 x 2048
    float* ybuf  = (float*)(ws + (size_t)MROWS * (XZ_N + 2 * D_INNER + XD_N) * 4);// 2048 x 2048

    const dim3 blk(256);

    // 1) xz = x @ W_in   (M=2048, K=1024, N=4096)
    {
        const int ng = XZ_N / 64;
        gemm_wmma_f32<2, 4, 0><<<gemm_grid(MROWS, ng), blk, 0, stream>>>(
            x, W_in, xz, nullptr, MROWS, XZ_N, D_MODEL, D_MODEL, XZ_N, XZ_N, ng);
    }

    // 2) u_c = silu(conv(u) + b)
    {
        const int n = BATCH * SEQ * D_INNER;
        conv_silu_kernel<<<(n + 255) / 256, blk, 0, stream>>>(xz, conv_w, conv_b, u_c);
    }

    // 3) x_dbl = u_c @ W_x   (M=2048, K=2048, N=96) — one wave covers all 96 cols
    {
        gemm_wmma_f32<2, 6, 0><<<gemm_grid(MROWS, 1), blk, 0, stream>>>(
            u_c, W_x, x_dbl, nullptr, MROWS, XD_N, D_INNER, D_INNER, XD_N, XD_N, 1);
    }

    // 4) dt = softplus(x_dbl[:, :64] @ W_dt + b_dt)  (M=2048, K=64, N=2048)
    {
        const int ng = D_INNER / 64;
        gemm_wmma_f32<2, 4, 1><<<gemm_grid(MROWS, ng), blk, 0, stream>>>(
            x_dbl, W_dt, dt, b_dt, MROWS, D_INNER, DT_RANK, XD_N, D_INNER, D_INNER, ng);
    }

    // 5) selective scan + gating (16 lanes per channel)
    {
        const int n = BATCH * D_INNER * D_STATE;   // 65536 threads
        scan_kernel<<<(n + 255) / 256, blk, 0, stream>>>(
            xz, u_c, x_dbl, dt, A_log, Dvec, ybuf);
    }

    // 6) out = y @ W_out   (M=2048, K=2048, N=1024)
    {
        const int ng = D_MODEL / 64;
        gemm_wmma_f32<2, 4, 0><<<gemm_grid(MROWS, ng), blk, 0, stream>>>(
            ybuf, W_out, out, nullptr, MROWS, D_MODEL, D_INNER, D_INNER, D_MODEL, D_MODEL, ng);
    }
}